// GCNEncoder_77214922048129
// MI455X (gfx1250) — compile-verified
//
#include <hip/hip_runtime.h>

typedef __attribute__((ext_vector_type(2))) float v2f;
typedef __attribute__((ext_vector_type(8))) float v8f;

#define N_GRAPHS 64

// ---------------- degree / dinv ----------------

__global__ void k_degree(const int* __restrict__ col, unsigned* __restrict__ deg, int E) {
  int e = blockIdx.x * blockDim.x + threadIdx.x;
  if (e < E) atomicAdd(&deg[col[e]], 1u);
}

__global__ void k_dinv(const unsigned* __restrict__ deg, float* __restrict__ dinv, int n) {
  int i = blockIdx.x * blockDim.x + threadIdx.x;
  if (i < n) dinv[i] = rsqrtf((float)(deg[i] + 1u));  // +1 self-loop
}

// ---------------- WMMA f32 GEMM: out[r, :] = (A[r, :] @ W) * dinv[r] ----------------
// Block = 256 threads = 8 waves; each wave computes a 16-row strip, full OUT width.
// A-frag (16x4 f32, ISA 7.12.2): lane = 16*h + m, VGPR v <-> K = v + 2h  -> float2 load.
// B-frag (4x16 f32): VGPR v <-> K = v + 2h, lane m <-> N.
// D (16x16 f32): VGPR v <-> M = v + 8h, lane m <-> N.

template <int IN, int OUT>
__global__ __launch_bounds__(256) void k_gemm_scaled(const float* __restrict__ A,
                                                     const float* __restrict__ W,
                                                     const float* __restrict__ dinv,
                                                     float* __restrict__ out, int nRows) {
  constexpr int NT = OUT / 16;
  const int lane = threadIdx.x & 31;
  const int wave = threadIdx.x >> 5;
  const int h = lane >> 4;   // lane half
  const int m = lane & 15;
  const int rowBase = blockIdx.x * 128 + wave * 16;

  v8f acc[NT] = {};

  int arow = rowBase + m;
  if (arow > nRows - 1) arow = nRows - 1;  // clamp: bogus rows never stored
  const float* aptr = A + (size_t)arow * IN + 2 * h;

  for (int kb = 0; kb < IN; kb += 4) {
    float2 a2 = *(const float2*)(aptr + kb);
    v2f av; av.x = a2.x; av.y = a2.y;
    const float* w0 = W + (size_t)(kb + 2 * h) * OUT + m;
#pragma unroll
    for (int t = 0; t < NT; t++) {
      v2f bv; bv.x = w0[t * 16]; bv.y = w0[OUT + t * 16];
      acc[t] = __builtin_amdgcn_wmma_f32_16x16x4_f32(
          /*neg_a=*/false, av, /*neg_b=*/false, bv,
          /*c_mod=*/(short)0, acc[t], /*reuse_a=*/false, /*reuse_b=*/false);
    }
  }

#pragma unroll
  for (int v = 0; v < 8; v++) {
    int r = rowBase + v + 8 * h;
    if (r < nRows) {
      float s = dinv[r];
#pragma unroll
      for (int t = 0; t < NT; t++) out[(size_t)r * OUT + t * 16 + m] = acc[t][v] * s;
    }
  }
}

// ---------------- edge scatter: dst[col] += src[row] (D floats / edge) ----------------
// One thread per (edge, 4-float chunk): coalesced float4 gather + 4 f32 atomics (L2-resident).

template <int D>
__global__ void k_scatter(const float* __restrict__ src, const int* __restrict__ row,
                          const int* __restrict__ col, float* __restrict__ dst, int E) {
  constexpr int C = D / 4;
  long long idx = (long long)blockIdx.x * blockDim.x + threadIdx.x;
  if (idx >= (long long)E * C) return;
  int e = (int)(idx / C);
  int c = (int)(idx % C);
  int r = row[e];
  int cl = col[e];
  float4 v = *(const float4*)(src + (size_t)r * D + c * 4);
  float* d = dst + (size_t)cl * D + c * 4;
  atomicAdd(d + 0, v.x);
  atomicAdd(d + 1, v.y);
  atomicAdd(d + 2, v.z);
  atomicAdd(d + 3, v.w);
}

// ---------------- epilogue 1: h1 = relu(LN((agg + selfloop) * dinv + b1)) ----------------
// One wave per node; 128 dims -> 4 per lane; shfl_xor tree reductions. Writes over xws.

__global__ void k_epilogue1(const float* __restrict__ agg, float* __restrict__ xws,
                            const float* __restrict__ dinv, const float* __restrict__ b1,
                            const float* __restrict__ lnw, const float* __restrict__ lnb,
                            int n) {
  int lane = threadIdx.x & 31;
  int wave = threadIdx.x >> 5;
  int i = blockIdx.x * 8 + wave;
  if (i >= n) return;
  float di = dinv[i];
  float t[4];
  float s = 0.f;
#pragma unroll
  for (int j = 0; j < 4; j++) {
    int d = lane + 32 * j;
    t[j] = (agg[(size_t)i * 128 + d] + xws[(size_t)i * 128 + d]) * di + b1[d];
    s += t[j];
  }
#pragma unroll
  for (int o = 16; o > 0; o >>= 1) s += __shfl_xor(s, o, 32);
  float mu = s * (1.f / 128.f);
  float var = 0.f;
#pragma unroll
  for (int j = 0; j < 4; j++) { float d0 = t[j] - mu; var += d0 * d0; }
#pragma unroll
  for (int o = 16; o > 0; o >>= 1) var += __shfl_xor(var, o, 32);
  float inv = rsqrtf(var * (1.f / 128.f) + 1e-5f);
#pragma unroll
  for (int j = 0; j < 4; j++) {
    int d = lane + 32 * j;
    float o = (t[j] - mu) * inv * lnw[d] + lnb[d];
    xws[(size_t)i * 128 + d] = fmaxf(o, 0.f);
  }
}

// ---------------- epilogue 2: h = (agg + selfloop)*dinv + b2 ; pool atomics ----------------

__global__ void k_epilogue2(float* __restrict__ hout, const float* __restrict__ xw2s,
                            const float* __restrict__ dinv, const float* __restrict__ b2,
                            const int* __restrict__ batch, float* __restrict__ psum,
                            float* __restrict__ pcnt, int n) {
  int lane = threadIdx.x & 31;
  int wave = threadIdx.x >> 5;
  int i = blockIdx.x * 8 + wave;
  if (i >= n) return;
  float di = dinv[i];
  int b = batch[i];
#pragma unroll
  for (int j = 0; j < 2; j++) {
    int d = lane + 32 * j;
    float h = (hout[(size_t)i * 64 + d] + xw2s[(size_t)i * 64 + d]) * di + b2[d];
    hout[(size_t)i * 64 + d] = h;
    atomicAdd(&psum[b * 64 + d], h);
  }
  if (lane == 0) atomicAdd(&pcnt[b], 1.f);
}

__global__ void k_pool(const float* __restrict__ psum, const float* __restrict__ pcnt,
                       float* __restrict__ g) {
  int idx = blockIdx.x * blockDim.x + threadIdx.x;
  if (idx < N_GRAPHS * 64) {
    int gi = idx >> 6;
    g[idx] = psum[idx] / fmaxf(pcnt[gi], 1.f);
  }
}

// ---------------- launch ----------------

extern "C" void kernel_launch(void* const* d_in, const int* in_sizes, int n_in,
                              void* d_out, int out_size, void* d_ws, size_t ws_size,
                              hipStream_t stream) {
  const float* x   = (const float*)d_in[0];
  const int*   ei  = (const int*)d_in[1];
  const int*  batch= (const int*)d_in[2];
  const float* W1  = (const float*)d_in[3];
  const float* b1  = (const float*)d_in[4];
  const float* lnw = (const float*)d_in[5];
  const float* lnb = (const float*)d_in[6];
  const float* W2  = (const float*)d_in[7];
  const float* b2  = (const float*)d_in[8];

  const int n = in_sizes[0] / 64;   // 100000 nodes
  const int E = in_sizes[1] / 2;    // 1600000 edges
  const int* row = ei;              // edge_index[0]
  const int* col = ei + E;          // edge_index[1]

  // workspace layout (floats):
  // [deg n(u32)] [agg1 n*128] [psum 4096] [pcnt 64] | [dinv n] [xws n*128]
  float* ws = (float*)d_ws;
  unsigned* deg = (unsigned*)ws;
  float* agg1 = ws + n;                          // also reused as xw2s
  float* psum = agg1 + (size_t)n * 128;
  float* pcnt = psum + N_GRAPHS * 64;
  float* dinv = pcnt + N_GRAPHS;
  float* xws  = dinv + n;                        // also reused as h1

  float* hout = (float*)d_out;                   // n*64
  float* gout = hout + (size_t)n * 64;           // 64*64

  // zero: deg + agg1 + psum + pcnt (one contiguous memset), and h region of d_out
  size_t zbytes = ((size_t)n + (size_t)n * 128 + N_GRAPHS * 64 + N_GRAPHS) * sizeof(float);
  hipMemsetAsync(d_ws, 0, zbytes, stream);
  hipMemsetAsync(d_out, 0, (size_t)n * 64 * sizeof(float), stream);

  k_degree<<<(E + 255) / 256, 256, 0, stream>>>(col, deg, E);
  k_dinv<<<(n + 255) / 256, 256, 0, stream>>>(deg, dinv, n);

  // layer 1
  k_gemm_scaled<64, 128><<<(n + 127) / 128, 256, 0, stream>>>(x, W1, dinv, xws, n);
  {
    long long tot = (long long)E * 32;  // 128/4 chunks per edge
    k_scatter<128><<<(unsigned)((tot + 255) / 256), 256, 0, stream>>>(xws, row, col, agg1, E);
  }
  k_epilogue1<<<(n + 7) / 8, 256, 0, stream>>>(agg1, xws, dinv, b1, lnw, lnb, n);

  // layer 2 (xw2s reuses agg1; scatter accumulates straight into d_out)
  float* xw2s = agg1;
  k_gemm_scaled<128, 64><<<(n + 127) / 128, 256, 0, stream>>>(xws, W2, dinv, xw2s, n);
  {
    long long tot = (long long)E * 16;  // 64/4 chunks per edge
    k_scatter<64><<<(unsigned)((tot + 255) / 256), 256, 0, stream>>>(xw2s, row, col, hout, E);
  }
  k_epilogue2<<<(n + 7) / 8, 256, 0, stream>>>(hout, xw2s, dinv, b2, batch, psum, pcnt, n);
  k_pool<<<(N_GRAPHS * 64 + 255) / 256, 256, 0, stream>>>(psum, pcnt, gout);
}